// NeuroODE_68642167325299
// MI455X (gfx1250) — compile-verified
//
#include <hip/hip_runtime.h>
#include <hip/hip_bf16.h>

typedef __attribute__((ext_vector_type(2))) float v2f;
typedef __attribute__((ext_vector_type(8))) float v8f;

#define NTH 1024   // 32 wave32 waves, one WGP

// Damped-oscillator Euler integration as a parallel prefix-product of 2x2
// step matrices. Scan combiner runs on V_WMMA_F32_16X16X4_F32 (8 packed
// 2x2 products per instruction, block-diagonal layout). Phase-2 is fully
// branchless (cndmask-only) except the final guarded LDS scatter.
__global__ __launch_bounds__(NTH)
void NeuroODE_scan_kernel(const float* __restrict__ y0v,
                          const float* __restrict__ tv,
                          const float* __restrict__ wv,
                          const int*   __restrict__ stepn_p,
                          float*       __restrict__ out,
                          int nsamp)
{
    const int tid    = threadIdx.x;
    const int lane   = tid & 31;
    const int wave   = tid >> 5;
    const int nint   = nsamp - 1;                    // 8191 intervals
    const int chunk  = (nint + NTH - 1) / NTH;       // 8 per thread
    const int stepn  = *stepn_p;
    const float omega = wv[0];
    const float zeta  = wv[1];
    const float k0 = omega * omega;                  // omega^2
    const float k1 = 2.0f * zeta * omega;            // 2*zeta*omega

    // Ping-pong scan state: 1024 matrices x 4 floats x 2 buffers = 32 KB LDS
    __shared__ float X[2][NTH * 4];

    // ---------------- Phase 1: per-thread chunk transfer matrix ----------
    // Evolve the two columns of the identity with the reference's exact
    // Euler update order: a' = a + h*b ; b' = b + h*(-w^2*a - 2*z*w*b)
    float g00 = 1.0f, g01 = 0.0f, g10 = 0.0f, g11 = 1.0f;
    const int base = tid * chunk;
    for (int j = 0; j < chunk; ++j) {
        const int i = base + j;
        if (i < nint) {
            const float h = (tv[i + 1] - tv[i]) / (float)stepn;
            for (int s = 0; s < stepn; ++s) {
                const float na0 = g00 + h * g10;
                const float nb0 = g10 + h * (-k0 * g00 - k1 * g10);
                const float na1 = g01 + h * g11;
                const float nb1 = g11 + h * (-k0 * g01 - k1 * g11);
                g00 = na0; g10 = nb0; g01 = na1; g11 = nb1;
            }
        }
    }
    // row-major {m00,m01,m10,m11}
    X[0][tid * 4 + 0] = g00;
    X[0][tid * 4 + 1] = g01;
    X[0][tid * 4 + 2] = g10;
    X[0][tid * 4 + 3] = g11;
    __syncthreads();

    // ---------------- Phase 2: Hillis-Steele scan, WMMA combiner ---------
    // Level d: X_t <- X_t * X_{t-d}  (identity for t < d).
    // Pack 8 pair-products per v_wmma_f32_16x16x4_f32:
    //   A[2b+r, 0:2] = L_b row r   (lanes 0-15; lanes 16-31 = K2,K3 = 0)
    //   B[0:2, 2b+c] = R_b col c   (lanes 0-15 hold K rows 0,1; 16-31 = 0)
    //   D[2b+r, 2b+c] = (L_b R_b)[r,c]  -> read only diagonal 2x2 blocks.
    // Per-lane derived indices (uniform per level):
    const int  bl    = (lane & 15) >> 1;   // block 0..7
    const int  rc    = lane & 1;           // A-row / B-col within block
    const bool lo16  = lane < 16;
    const bool writer = (lane < 8) || (lane >= 24);
    const float idv0 = (rc == 0) ? 1.0f : 0.0f;   // identity col c, row 0
    const float idv1 = (rc == 1) ? 1.0f : 0.0f;   // identity col c, row 1
    // q0 in {0,2,4,6}: first of the two D-vgprs this writer lane scatters
    const int  wsrc  = ((lane < 8) ? lane : (lane - 24)) & 6;
    const bool qb2   = (wsrc & 2) != 0;
    const bool qb4   = (wsrc & 4) != 0;

    int cur = 0;
#pragma unroll 1
    for (int dlev = 1; dlev < NTH; dlev <<= 1) {
        const float* __restrict__ src = X[cur];
        float* __restrict__ dst = X[cur ^ 1];
#pragma unroll
        for (int k = 0; k < 4; ++k) {           // 4 WMMAs/wave -> 32 slots
            const int tA = wave * 32 + k * 8 + bl;
            // A operand (unconditional loads, masked by cndmask)
            const float axl = src[tA * 4 + rc * 2 + 0];
            const float ayl = src[tA * 4 + rc * 2 + 1];
            // B operand: X_{t-d} or identity (clamped address, then select)
            const int  sIdx = tA - dlev;
            const int  sCl  = (sIdx < 0) ? 0 : sIdx;
            const float bxl = src[sCl * 4 + 0 + rc];   // R[0,c]
            const float byl = src[sCl * 4 + 2 + rc];   // R[1,c]
            const bool hasL = sIdx >= 0;

            v2f a, b;
            a.x = lo16 ? axl : 0.0f;
            a.y = lo16 ? ayl : 0.0f;
            b.x = lo16 ? (hasL ? bxl : idv0) : 0.0f;
            b.y = lo16 ? (hasL ? byl : idv1) : 0.0f;

            v8f cacc = {};
            v8f dres = __builtin_amdgcn_wmma_f32_16x16x4_f32(
                false, a, false, b, (short)0, cacc, false, false);

            // Branchless extraction of the two diagonal-block entries
            // (vgprs q0 and q0+1), computed by all lanes; stored by writers.
            const float s02 = qb2 ? dres[2] : dres[0];
            const float s46 = qb2 ? dres[6] : dres[4];
            const float e0  = qb4 ? s46 : s02;          // row 0 of block
            const float s13 = qb2 ? dres[3] : dres[1];
            const float s57 = qb2 ? dres[7] : dres[5];
            const float e1  = qb4 ? s57 : s13;          // row 1 of block
            if (writer) {
                dst[tA * 4 + 0 + rc] = e0;
                dst[tA * 4 + 2 + rc] = e1;
            }
        }
        __syncthreads();
        cur ^= 1;
    }

    // ---------------- Phase 3: apply exclusive prefix, emit rows ---------
    float p00, p01, p10, p11;
    if (tid == 0) {
        p00 = 1.0f; p01 = 0.0f; p10 = 0.0f; p11 = 1.0f;
    } else {
        const float* S = &X[cur][(tid - 1) * 4];
        p00 = S[0]; p01 = S[1]; p10 = S[2]; p11 = S[3];
    }
    const float y0 = y0v[0];
    const float y1 = y0v[1];
    float v0 = p00 * y0 + p01 * y1;
    float v1 = p10 * y0 + p11 * y1;
    if (tid == 0) {                 // first output row is y0 itself
        *(float2*)&out[0] = make_float2(y0, y1);
    }
    for (int j = 0; j < chunk; ++j) {
        const int i = base + j;
        if (i < nint) {
            const float h = (tv[i + 1] - tv[i]) / (float)stepn;
            for (int s = 0; s < stepn; ++s) {
                const float d0 = v1;
                const float d1 = -k0 * v0 - k1 * v1;
                v0 = v0 + h * d0;
                v1 = v1 + h * d1;
            }
            *(float2*)&out[(i + 1) * 2] = make_float2(v0, v1);
        }
    }
}

extern "C" void kernel_launch(void* const* d_in, const int* in_sizes, int n_in,
                              void* d_out, int out_size, void* d_ws, size_t ws_size,
                              hipStream_t stream) {
    const float* y0 = (const float*)d_in[0];   // (2,) f32
    const float* t  = (const float*)d_in[1];   // (8192,) f32
    const float* w  = (const float*)d_in[2];   // (2,) f32  [omega, zeta]
    const int* sn   = (const int*)d_in[3];     // scalar step_n
    float* out      = (float*)d_out;           // (8192, 2) f32
    const int nsamp = in_sizes[1];
    (void)n_in; (void)out_size; (void)d_ws; (void)ws_size;

    NeuroODE_scan_kernel<<<1, NTH, 0, stream>>>(y0, t, w, sn, out, nsamp);
}